// CMCAccuracy_50268297232554
// MI455X (gfx1250) — compile-verified
//
#include <hip/hip_runtime.h>
#include <hip/hip_bf16.h>

#define NN 16384
#define DD 256
#define TOPK 5
#define ROWS 32          // rows per block
#define LDA 260          // padded LDS row stride (dwords) for A tile
#define LDD 68           // padded LDS row stride for distance tile
#define NCT (NN / 64)    // 256 column tiles of 64
#define PANEL (DD * 64)  // B panel: 256 k x 64 cols (floats)

typedef __attribute__((ext_vector_type(2))) float v2f;
typedef __attribute__((ext_vector_type(8))) float v8f;
typedef __attribute__((ext_vector_type(4))) unsigned int u32x4;
typedef __attribute__((ext_vector_type(4))) int i32x4;
typedef __attribute__((ext_vector_type(8))) int i32x8;

#ifndef __has_builtin
#define __has_builtin(x) 0
#endif
#if __has_builtin(__builtin_amdgcn_tensor_load_to_lds)
#define HAVE_TDM 1
#else
#define HAVE_TDM 0
#endif

// dynamic LDS layout (bytes): panels | A | dist | merge-d | merge-c
#define SMEM_BYTES ((2 * PANEL + ROWS * LDA + ROWS * LDD + ROWS * 40) * 4 + ROWS * 40 * 4)

// ---------------- kernel 1: row squared norms ----------------
__global__ void cmc_norms_kernel(const float* __restrict__ e, float* __restrict__ s) {
    int i = blockIdx.x * blockDim.x + threadIdx.x;
    const float4* row = (const float4*)e + (size_t)i * (DD / 4);
    float acc = 0.0f;
#pragma unroll
    for (int q = 0; q < DD / 4; ++q) {
        float4 v = row[q];
        acc += v.x * v.x + v.y * v.y + v.z * v.z + v.w * v.w;
    }
    s[i] = acc;
}

// ---------------- kernel 2: transpose E -> E^T [DD][NN] ----------------
__global__ void cmc_transpose_kernel(const float* __restrict__ e, float* __restrict__ eT) {
    __shared__ float tile[32][33];
    const int i0 = blockIdx.x * 32;
    const int k0 = blockIdx.y * 32;
    const int tx = threadIdx.x, ty = threadIdx.y;
#pragma unroll
    for (int r = ty; r < 32; r += 8)
        tile[r][tx] = e[(size_t)(i0 + r) * DD + (k0 + tx)];
    __syncthreads();
#pragma unroll
    for (int r = ty; r < 32; r += 8)
        eT[(size_t)(k0 + r) * NN + (i0 + tx)] = tile[tx][r];
}

// ---------------- kernel 3: zero the hit counter ----------------
__global__ void cmc_init_kernel(unsigned int* cnt) {
    if (threadIdx.x == 0) *cnt = 0u;
}

// ---------------- kernel 4: TDM-fed fp32-WMMA Gram + top-5 + label match ----
// 256 threads = 8 waves = 2 row-tiles x 4 col-subtiles. Block owns 32 rows.
// B panels (64 cols x 256 k) DMA'd into double-buffered LDS by the TDM.
__global__ __launch_bounds__(256) void cmc_gram_topk_kernel(
    const float* __restrict__ e, const float* __restrict__ eT,
    const float* __restrict__ s, const int* __restrict__ labels,
    unsigned int* __restrict__ cnt) {

    extern __shared__ char smem_raw[];
    float* lds_b    = (float*)smem_raw;           // 2 * PANEL floats
    float* lds_a    = lds_b + 2 * PANEL;          // ROWS * LDA
    float* lds_dist = lds_a + ROWS * LDA;         // ROWS * LDD
    float* lds_md   = lds_dist + ROWS * LDD;      // ROWS * 40
    int*   lds_mc   = (int*)(lds_md + ROWS * 40); // ROWS * 40

    const int tid  = threadIdx.x;
    const int lane = tid & 31;
    const int wid  = tid >> 5;            // wave 0..7
    const int widc = wid & 3;             // col sub-tile 0..3
    const int rt0  = (wid >> 2) * 16;     // row tile base 0 or 16
    const int mm   = lane & 15;           // M (A) / N (B,C) index
    const int ko   = (lane >> 4) << 1;    // K offset 0 or 2
    const int row0 = blockIdx.x * ROWS;

    // ---- stage A tile (32 x 256) into LDS ----
    {
        const float4* e4 = (const float4*)e;
#pragma unroll
        for (int it = 0; it < 8; ++it) {
            int idx = tid + it * 256;             // 0..2047 float4 chunks
            int m = idx >> 6;
            int q = idx & 63;
            float4 v = e4[(size_t)(row0 + m) * (DD / 4) + q];
            float* dst = &lds_a[m * LDA + q * 4];
            dst[0] = v.x; dst[1] = v.y; dst[2] = v.z; dst[3] = v.w;
        }
    }
    __syncthreads();

    const int rbase = (lane < 16) ? 0 : 8;
    float sr[8];
#pragma unroll
    for (int v = 0; v < 8; ++v) sr[v] = s[row0 + rt0 + rbase + v];

    const int selr = tid & 31;            // selection row 0..31
    const int selc = tid >> 5;            // column chunk 0..7
    float td[TOPK]; int tc[TOPK];
#pragma unroll
    for (int q = 0; q < TOPK; ++q) { td[q] = __builtin_inff(); tc[q] = -1; }

#if HAVE_TDM
    // Issue a 2-D TDM descriptor: 64 cols x 256 k of eT -> LDS panel.
    auto tdm_issue = [&](int ctn) {
        const unsigned lds_off =
            (unsigned)(unsigned long long)(lds_b + (ctn & 1) * PANEL);
        const unsigned long long ga = (unsigned long long)(eT + ctn * 64);
        u32x4 g0;
        g0.x = 1u;                                        // count=1, user mode
        g0.y = lds_off;                                   // LDS byte address
        g0.z = (unsigned)(ga & 0xFFFFFFFFull);            // global addr lo
        g0.w = (unsigned)((ga >> 32) & 0x01FFFFFFull)     // global addr hi
             | (2u << 30);                                // type = image
        i32x8 g1;
        g1[0] = 0x00020000;                               // data_size = 4 B
        g1[1] = (int)(((unsigned)NN & 0xFFFFu) << 16);    // tensor_dim0 lo16
        g1[2] = (int)((unsigned)NN >> 16)                 // tensor_dim0 hi16
              | (int)(((unsigned)DD & 0xFFFFu) << 16);    // tensor_dim1 lo16
        g1[3] = (int)((unsigned)DD >> 16)                 // tensor_dim1 hi16
              | (int)(64u << 16);                         // tile_dim0 = 64
        g1[4] = (int)DD;                                  // tile_dim1=256, tile_dim2=0
        g1[5] = (int)NN;                                  // tensor_dim0_stride lo32
        g1[6] = 0;
        g1[7] = 0;
        i32x4 gz = {0, 0, 0, 0};
#if __clang_major__ >= 23
        i32x8 gz8 = {0, 0, 0, 0, 0, 0, 0, 0};
        __builtin_amdgcn_tensor_load_to_lds(g0, g1, gz, gz, gz8, 0);
#else
        __builtin_amdgcn_tensor_load_to_lds(g0, g1, gz, gz, 0);
#endif
    };
#endif

    // Compute + select on one 32x64 tile fed from an LDS B panel.
    auto do_tile = [&](int ct, const float* __restrict__ bpanel) {
        const int c0 = ct * 64 + widc * 16;
        const float* bcol = bpanel + widc * 16 + mm;
        v8f acc = {};
#pragma unroll
        for (int k = 0; k < DD; k += 4) {
            // A 16x4 fragment: lane m holds {A[m][k+ko], A[m][k+ko+1]}
            const float2 af = *(const float2*)&lds_a[(rt0 + mm) * LDA + ko + k];
            v2f a = { af.x, af.y };
            // B 4x16 fragment from LDS panel (k-major, 64 floats per k row)
            v2f b = { bcol[(k + ko) * 64], bcol[(k + ko + 1) * 64] };
            acc = __builtin_amdgcn_wmma_f32_16x16x4_f32(
                false, a, false, b, (short)0, acc, false, false);
        }
        const float sc  = s[c0 + mm];
        const int   col = c0 + mm;
#pragma unroll
        for (int v = 0; v < 8; ++v) {
            const int rl = rt0 + rbase + v;               // local row 0..31
            float d = sr[v] + sc - 2.0f * acc[v];
            if (col == row0 + rl) d = __builtin_inff();   // exclude self
            lds_dist[rl * LDD + widc * 16 + mm] = d;
        }
        __syncthreads();
#pragma unroll
        for (int j = 0; j < 8; ++j) {
            const int jj = selc * 8 + j;
            const float d = lds_dist[selr * LDD + jj];
            if (d < td[TOPK - 1]) {
                const int c = ct * 64 + jj;
#pragma unroll
                for (int p = TOPK - 1; p >= 1; --p) {
                    const bool shift = d < td[p - 1];
                    const bool here  = !shift && (d < td[p]);
                    float nd = shift ? td[p - 1] : (here ? d : td[p]);
                    int   nc = shift ? tc[p - 1] : (here ? c : tc[p]);
                    td[p] = nd; tc[p] = nc;
                }
                if (d < td[0]) { td[0] = d; tc[0] = c; }
            }
        }
        __syncthreads();
    };

#if HAVE_TDM
    if (wid == 0) tdm_issue(0);
    for (int ct = 0; ct < NCT - 1; ++ct) {
        if (wid == 0) {
            tdm_issue(ct + 1);                     // prefetch next panel
            __builtin_amdgcn_s_wait_tensorcnt(1);  // panel ct has landed
        }
        __syncthreads();
        do_tile(ct, lds_b + (ct & 1) * PANEL);
    }
    if (wid == 0) __builtin_amdgcn_s_wait_tensorcnt(0);
    __syncthreads();
    do_tile(NCT - 1, lds_b + ((NCT - 1) & 1) * PANEL);
#else
    for (int ct = 0; ct < NCT; ++ct) {
        // cooperative panel copy fallback (single buffer)
        for (int idx = tid; idx < DD * 16; idx += 256) {
            int k = idx >> 4;
            int q = idx & 15;
            float4 v = *(const float4*)(eT + (size_t)k * NN + ct * 64 + q * 4);
            float* dst = &lds_b[k * 64 + q * 4];
            dst[0] = v.x; dst[1] = v.y; dst[2] = v.z; dst[3] = v.w;
        }
        __syncthreads();
        do_tile(ct, lds_b);
    }
#endif

    // ---- merge 8 partial top-5 lists per row, compare labels, count ----
#pragma unroll
    for (int q = 0; q < TOPK; ++q) {
        lds_md[selr * 40 + selc * TOPK + q] = td[q];
        lds_mc[selr * 40 + selc * TOPK + q] = tc[q];
    }
    __syncthreads();

    if (tid < ROWS) {
        const int r = tid;
        float bd[TOPK]; int bc[TOPK];
#pragma unroll
        for (int q = 0; q < TOPK; ++q) { bd[q] = __builtin_inff(); bc[q] = -1; }
        for (int t = 0; t < 40; ++t) {
            const float d = lds_md[r * 40 + t];
            const int   c = lds_mc[r * 40 + t];
            if (d < bd[TOPK - 1]) {
#pragma unroll
                for (int p = TOPK - 1; p >= 1; --p) {
                    const bool shift = d < bd[p - 1];
                    const bool here  = !shift && (d < bd[p]);
                    float nd = shift ? bd[p - 1] : (here ? d : bd[p]);
                    int   nc = shift ? bc[p - 1] : (here ? c : bc[p]);
                    bd[p] = nd; bc[p] = nc;
                }
                if (d < bd[0]) { bd[0] = d; bc[0] = c; }
            }
        }
        const int L = labels[row0 + r];
        int hit = 0;
#pragma unroll
        for (int q = 0; q < TOPK; ++q)
            if (bc[q] >= 0 && labels[bc[q]] == L) hit = 1;
        atomicAdd(cnt, (unsigned int)hit);
    }
}

// ---------------- kernel 5: finalize ----------------
__global__ void cmc_finalize_kernel(const unsigned int* cnt, float* out) {
    if (threadIdx.x == 0) out[0] = (float)(*cnt) * (1.0f / (float)NN);
}

extern "C" void kernel_launch(void* const* d_in, const int* in_sizes, int n_in,
                              void* d_out, int out_size, void* d_ws, size_t ws_size,
                              hipStream_t stream) {
    const float* e      = (const float*)d_in[0];   // [NN, DD] fp32
    const int*   labels = (const int*)d_in[1];     // [NN] int
    float* out = (float*)d_out;                    // scalar fp32

    // workspace layout: s[NN] | eT[DD*NN] | counter  (~16.1 MB)
    float* s  = (float*)d_ws;
    float* eT = s + NN;
    unsigned int* cnt = (unsigned int*)(eT + (size_t)NN * DD);

    cmc_norms_kernel<<<NN / 256, 256, 0, stream>>>(e, s);

    dim3 tb(32, 8);
    dim3 tg(NN / 32, DD / 32);
    cmc_transpose_kernel<<<tg, tb, 0, stream>>>(e, eT);

    cmc_init_kernel<<<1, 32, 0, stream>>>(cnt);

    cmc_gram_topk_kernel<<<NN / ROWS, 256, SMEM_BYTES, stream>>>(e, eT, s, labels, cnt);

    cmc_finalize_kernel<<<1, 32, 0, stream>>>(cnt, out);
}